// HierMatcher_20306605376085
// MI455X (gfx1250) — compile-verified
//
#include <hip/hip_runtime.h>
#include <hip/hip_bf16.h>

// Problem constants (from reference)
#define BB   256
#define FF   4
#define LL   64
#define DD   768
#define IN1  (2 * FF * DD)   // 6144
#define HID1 1600
#define NEGV (-1e10f)

typedef __attribute__((ext_vector_type(2))) float v2f;
typedef __attribute__((ext_vector_type(8))) float v8f;

// ---------------------------------------------------------------------------
// Kernel 1: masked attention pooling, writes directly into concat [B, 6144].
// One block per (b, side, f). 2048 blocks x 256 threads.
// ---------------------------------------------------------------------------
__global__ __launch_bounds__(256) void attn_pool_kernel(
    const float* __restrict__ left, const float* __restrict__ right,
    const float* __restrict__ femb_l, const float* __restrict__ femb_r,
    const float* __restrict__ empty_attr, float* __restrict__ concat) {
  __shared__ float s_part[256];
  __shared__ int   s_nzp[256];
  __shared__ float s_logit[LL];
  __shared__ int   s_tok[LL];
  __shared__ float s_w[LL];
  __shared__ float s_red[LL];
  __shared__ int   s_attr;

  const int g    = blockIdx.x;
  const int f    = g & 3;
  const int side = (g >> 2) & 1;
  const int b    = g >> 3;

  const float* v    = (side == 0 ? left : right) + (size_t)(b * FF + f) * LL * DD;
  const float* femb = (side == 0 ? femb_l : femb_r) + (size_t)f * DD;
  float* outp = concat + (size_t)b * IN1 + (size_t)(side * FF + f) * DD;

  const int tid = threadIdx.x;

  // ---- Pass 1: logits[l] = dot(v[l,:], femb) and token nonzero mask --------
  {
    const int row = tid >> 2;    // 64 rows, 4 threads per row
    const int sub = tid & 3;
    const float* vr = v + (size_t)row * DD;
    float acc = 0.0f;
    int nz = 0;
    for (int d = sub; d < DD; d += 4) {
      float x = vr[d];
      acc += x * femb[d];
      nz |= (x != 0.0f) ? 1 : 0;
    }
    s_part[tid] = acc;
    s_nzp[tid]  = nz;
  }
  __syncthreads();
  if ((tid & 3) == 0) {
    const int row = tid >> 2;
    float lg = s_part[tid] + s_part[tid + 1] + s_part[tid + 2] + s_part[tid + 3];
    int   m  = s_nzp[tid] | s_nzp[tid + 1] | s_nzp[tid + 2] | s_nzp[tid + 3];
    s_logit[row] = m ? lg : (lg + NEGV);   // masked tokens get NEG added
    s_tok[row]   = m;
  }
  __syncthreads();

  // ---- Masked softmax over the 64 tokens (LDS tree reductions) -------------
  if (tid < LL) s_red[tid] = s_logit[tid];
  __syncthreads();
  for (int s = 32; s >= 1; s >>= 1) {
    if (tid < s) s_red[tid] = fmaxf(s_red[tid], s_red[tid + s]);
    __syncthreads();
  }
  const float mx = s_red[0];
  __syncthreads();
  if (tid < LL) {
    float e = __expf(s_logit[tid] - mx);
    s_w[tid]   = e;
    s_red[tid] = e;
  }
  __syncthreads();
  for (int s = 32; s >= 1; s >>= 1) {
    if (tid < s) s_red[tid] += s_red[tid + s];
    __syncthreads();
  }
  const float denom = s_red[0];
  if (tid < LL) s_w[tid] = s_w[tid] / denom;
  if (tid == 0) {
    int a = 0;
    #pragma unroll
    for (int i = 0; i < LL; ++i) a |= s_tok[i];
    s_attr = a;
  }
  __syncthreads();

  // ---- Pass 2: summary[d] = sum_l w[l] * v[l,d] (coalesced in d) -----------
  if (s_attr) {
    for (int d = tid; d < DD; d += 256) {
      const float* col = v + d;
      float acc = 0.0f;
      #pragma unroll 8
      for (int l = 0; l < LL; ++l) acc += s_w[l] * col[(size_t)l * DD];
      outp[d] = acc;
    }
  } else {
    for (int d = tid; d < DD; d += 256) outp[d] = empty_attr[d];
  }
}

// ---------------------------------------------------------------------------
// Kernel 2: H = relu(concat @ W1^T + b1), fp32 WMMA 16x16x4.
// M=256, N=1600, K=6144. One wave per 16x16 tile; 1600 tiles = 200 blocks x 8
// waves. A fragment: lane half selects K pair (0,1)/(2,3); row = lane&15.
// B fragment: same pattern reading W1 row-major (W1^T columns are W1 rows).
// ---------------------------------------------------------------------------
__global__ __launch_bounds__(256) void gemm1_wmma_kernel(
    const float* __restrict__ A,   // concat [256, 6144]
    const float* __restrict__ W1,  // [1600, 6144]
    const float* __restrict__ b1,  // [1600]
    float* __restrict__ H) {       // [256, 1600]
  const int waveId = threadIdx.x >> 5;
  const int lane   = threadIdx.x & 31;
  const int tile   = blockIdx.x * 8 + waveId;
  const int tileM  = tile / 100;             // 0..15
  const int tileN  = tile - tileM * 100;     // 0..99
  const int half   = lane >> 4;              // 0: K={0,1}  1: K={2,3}
  const int lid    = lane & 15;

  const float* aPtr = A  + (size_t)(tileM * 16 + lid) * IN1 + half * 2;
  const float* bPtr = W1 + (size_t)(tileN * 16 + lid) * IN1 + half * 2;

  v8f acc = {};
  for (int k = 0; k < IN1; k += 16) {
    __builtin_prefetch(bPtr + k + 1024, 0, 1);  // global_prefetch_b8 ahead in K
    v2f a0 = *(const v2f*)(aPtr + k);
    v2f w0 = *(const v2f*)(bPtr + k);
    v2f a1 = *(const v2f*)(aPtr + k + 4);
    v2f w1 = *(const v2f*)(bPtr + k + 4);
    v2f a2 = *(const v2f*)(aPtr + k + 8);
    v2f w2 = *(const v2f*)(bPtr + k + 8);
    v2f a3 = *(const v2f*)(aPtr + k + 12);
    v2f w3 = *(const v2f*)(bPtr + k + 12);
    acc = __builtin_amdgcn_wmma_f32_16x16x4_f32(false, a0, false, w0, (short)0, acc, false, false);
    acc = __builtin_amdgcn_wmma_f32_16x16x4_f32(false, a1, false, w1, (short)0, acc, false, false);
    acc = __builtin_amdgcn_wmma_f32_16x16x4_f32(false, a2, false, w2, (short)0, acc, false, false);
    acc = __builtin_amdgcn_wmma_f32_16x16x4_f32(false, a3, false, w3, (short)0, acc, false, false);
  }

  // C/D layout: VGPR r holds row M=r (lanes 0-15) or M=r+8 (lanes 16-31),
  // column N = lane&15. Fuse bias + ReLU on store.
  const int   nCol  = tileN * 16 + lid;
  const float bias  = b1[nCol];
  const int   mBase = tileM * 16 + half * 8;
  float* hOut = H + (size_t)mBase * HID1 + nCol;
  #pragma unroll
  for (int r = 0; r < 8; ++r) {
    float val = acc[r] + bias;
    hOut[(size_t)r * HID1] = val > 0.0f ? val : 0.0f;
  }
}

// ---------------------------------------------------------------------------
// Kernel 3: out = H @ W2^T + b2   (tiny: 256x2)
// ---------------------------------------------------------------------------
__global__ __launch_bounds__(128) void gemm2_kernel(
    const float* __restrict__ H, const float* __restrict__ W2,
    const float* __restrict__ b2, float* __restrict__ out) {
  __shared__ float s0[128];
  __shared__ float s1[128];
  const int b   = blockIdx.x;
  const int tid = threadIdx.x;
  const float* h = H + (size_t)b * HID1;
  float a0 = 0.0f, a1 = 0.0f;
  for (int i = tid; i < HID1; i += 128) {
    float hv = h[i];
    a0 += hv * W2[i];
    a1 += hv * W2[HID1 + i];
  }
  s0[tid] = a0;
  s1[tid] = a1;
  __syncthreads();
  for (int s = 64; s >= 1; s >>= 1) {
    if (tid < s) { s0[tid] += s0[tid + s]; s1[tid] += s1[tid + s]; }
    __syncthreads();
  }
  if (tid == 0) {
    out[b * 2 + 0] = s0[0] + b2[0];
    out[b * 2 + 1] = s1[0] + b2[1];
  }
}

// ---------------------------------------------------------------------------
extern "C" void kernel_launch(void* const* d_in, const int* in_sizes, int n_in,
                              void* d_out, int out_size, void* d_ws, size_t ws_size,
                              hipStream_t stream) {
  (void)in_sizes; (void)n_in; (void)out_size; (void)ws_size;
  const float* left       = (const float*)d_in[0];
  const float* right      = (const float*)d_in[1];
  const float* femb_l     = (const float*)d_in[2];
  const float* femb_r     = (const float*)d_in[3];
  const float* empty_attr = (const float*)d_in[4];
  const float* W1         = (const float*)d_in[5];
  const float* b1         = (const float*)d_in[6];
  const float* W2         = (const float*)d_in[7];
  const float* b2         = (const float*)d_in[8];
  float* out = (float*)d_out;

  float* concat = (float*)d_ws;                      // [256, 6144]
  float* H      = concat + (size_t)BB * IN1;         // [256, 1600]

  attn_pool_kernel<<<BB * FF * 2, 256, 0, stream>>>(left, right, femb_l, femb_r,
                                                    empty_attr, concat);
  gemm1_wmma_kernel<<<200, 256, 0, stream>>>(concat, W1, b1, H);
  gemm2_kernel<<<BB, 128, 0, stream>>>(H, W2, b2, out);
}